// TransformerBlock_90091234001438
// MI455X (gfx1250) — compile-verified
//
#include <hip/hip_runtime.h>
#include <cmath>

typedef __attribute__((ext_vector_type(2))) float v2f;
typedef __attribute__((ext_vector_type(8))) float v8f;

static constexpr int kB     = 16;    // batch
static constexpr int kD     = 2048;  // model dim
static constexpr int kH     = 16;    // heads
static constexpr int kHD    = 128;   // head dim
static constexpr int kLP    = 2048;  // past length
static constexpr int kLT    = 2049;  // total keys
static constexpr int kM     = 8192;  // MLP dim

// ---------------------------------------------------------------------------
// LayerNorm: one block per row (16 rows of 2048)
// ---------------------------------------------------------------------------
__global__ __launch_bounds__(256) void ln_rows_kernel(
    const float* __restrict__ x, const float* __restrict__ w,
    const float* __restrict__ b, float* __restrict__ y) {
  __shared__ float rs[256], rs2[256];
  const int row = blockIdx.x;
  const int tid = threadIdx.x;
  const float* xr = x + (size_t)row * kD;
  float s = 0.f, s2 = 0.f;
  for (int i = tid; i < kD; i += 256) {
    float v = xr[i];
    s += v;
    s2 = fmaf(v, v, s2);
  }
  rs[tid] = s; rs2[tid] = s2;
  __syncthreads();
  for (int st = 128; st > 0; st >>= 1) {
    if (tid < st) { rs[tid] += rs[tid + st]; rs2[tid] += rs2[tid + st]; }
    __syncthreads();
  }
  const float mean = rs[0] * (1.f / kD);
  const float var  = rs2[0] * (1.f / kD) - mean * mean;
  const float rstd = rsqrtf(var + 1e-5f);
  float* yr = y + (size_t)row * kD;
  for (int i = tid; i < kD; i += 256)
    yr[i] = (xr[i] - mean) * rstd * w[i] + b[i];
}

// ---------------------------------------------------------------------------
// M=16 GEMM via V_WMMA_F32_16X16X4_F32:  out[16,N] = A[16,K] @ W[N,K]^T (+epi)
// One wave per 16-column tile; 4 waves per block.
// epi: 0 = none, 1 = +res, 2 = +bias then exact GELU, 3 = +bias +res
// ---------------------------------------------------------------------------
__global__ __launch_bounds__(128) void gemm16_wmma_kernel(
    const float* __restrict__ A, const float* __restrict__ W,
    const float* __restrict__ bias, const float* __restrict__ res,
    float* __restrict__ out, int K, int N, int epi) {
  const int lane = threadIdx.x & 31;
  const int wv   = threadIdx.x >> 5;
  const int n0   = (blockIdx.x * 4 + wv) * 16;
  const int m    = lane & 15;          // A row / W row-within-tile
  const int kh   = (lane >> 4) * 2;    // K-half select per ISA A/B layout

  const float2* ap = (const float2*)(A + (size_t)m * K + kh);
  const float2* bp = (const float2*)(W + (size_t)(n0 + m) * K + kh);

  v8f acc = {};
  const int kiter = K >> 2;
#pragma unroll 4
  for (int i = 0; i < kiter; ++i) {
    const float2 av = ap[2 * i];
    const float2 bv = bp[2 * i];
    v2f a;  a[0] = av.x;  a[1] = av.y;
    v2f bb; bb[0] = bv.x; bb[1] = bv.y;
    acc = __builtin_amdgcn_wmma_f32_16x16x4_f32(
        false, a, false, bb, (short)0, acc, false, false);
  }

  // C/D layout: VGPR v -> row v + 8*(lane/16), column lane%16
  const int crow0 = (lane >> 4) * 8;
  const int col   = n0 + m;
  const float bval = (epi >= 2) ? bias[col] : 0.f;
#pragma unroll
  for (int v = 0; v < 8; ++v) {
    const int r = crow0 + v;
    float val = acc[v];
    if (epi == 1) {
      val += res[(size_t)r * N + col];
    } else if (epi == 2) {
      val += bval;
      val = 0.5f * val * (1.f + erff(val * 0.70710678118654752440f));
    } else if (epi == 3) {
      val += bval + res[(size_t)r * N + col];
    }
    out[(size_t)r * N + col] = val;
  }
}

// ---------------------------------------------------------------------------
// Decode attention: one 8-wave block per (b,h); each wave streams every 8th
// key/value row (512B per row == one b128 load per lane), online softmax,
// cross-wave merge through LDS.
// ---------------------------------------------------------------------------
__global__ __launch_bounds__(256) void attn_kernel(
    const float* __restrict__ qkv, const float* __restrict__ pk,
    const float* __restrict__ pv, float* __restrict__ attn_out) {
  const int b    = blockIdx.x >> 4;
  const int h    = blockIdx.x & 15;
  const int lane = threadIdx.x & 31;
  const int wv   = threadIdx.x >> 5;  // 0..7

  __shared__ float m_s[8], s_s[8];
  __shared__ __align__(16) float acc_s[8 * kHD];

  const float scale = 0.08838834764831845f;  // 1/sqrt(128)
  const float* qp = qkv + (size_t)b * 3 * kD + h * kHD + lane * 4;
  float4 q = *(const float4*)qp;
  q.x *= scale; q.y *= scale; q.z *= scale; q.w *= scale;

  float mrun = -INFINITY, srun = 0.f;
  float4 acc = {0.f, 0.f, 0.f, 0.f};

  for (int l = wv; l < kLT; l += 8) {
    const float* kp;
    const float* vp;
    if (l < kLP) {
      const size_t off = (((size_t)b * kLP + l) * kH + h) * kHD;
      kp = pk + off;
      vp = pv + off;
    } else {  // current token's k/v from the qkv projection
      kp = qkv + (size_t)b * 3 * kD + kD + h * kHD;
      vp = qkv + (size_t)b * 3 * kD + 2 * kD + h * kHD;
    }
    const float4 kk = *(const float4*)(kp + lane * 4);
    float p = q.x * kk.x + q.y * kk.y + q.z * kk.z + q.w * kk.w;
#pragma unroll
    for (int off = 16; off > 0; off >>= 1) p += __shfl_xor(p, off, 32);

    const float mnew = fmaxf(mrun, p);
    const float c = __expf(mrun - mnew);
    const float e = __expf(p - mnew);
    srun = srun * c + e;
    const float4 vvv = *(const float4*)(vp + lane * 4);
    acc.x = acc.x * c + e * vvv.x;
    acc.y = acc.y * c + e * vvv.y;
    acc.z = acc.z * c + e * vvv.z;
    acc.w = acc.w * c + e * vvv.w;
    mrun = mnew;
  }

  if (lane == 0) { m_s[wv] = mrun; s_s[wv] = srun; }
  *(float4*)&acc_s[wv * kHD + lane * 4] = acc;
  __syncthreads();

  if (threadIdx.x < kHD) {
    const int d = threadIdx.x;
    float Mx = -INFINITY;
#pragma unroll
    for (int w = 0; w < 8; ++w) Mx = fmaxf(Mx, m_s[w]);
    float tot = 0.f, o = 0.f;
#pragma unroll
    for (int w = 0; w < 8; ++w) {
      const float f = __expf(m_s[w] - Mx);
      tot += f * s_s[w];
      o   += f * acc_s[w * kHD + d];
    }
    attn_out[(size_t)b * kD + h * kHD + d] = o / tot;
  }
}

// ---------------------------------------------------------------------------
// Launch: LN0 -> QKV gemm -> attention -> Wo gemm(+res) -> LN1 ->
//         FC0 gemm(+bias,gelu) -> FC1 gemm(+bias,+res) -> d_out
// ---------------------------------------------------------------------------
extern "C" void kernel_launch(void* const* d_in, const int* in_sizes, int n_in,
                              void* d_out, int out_size, void* d_ws, size_t ws_size,
                              hipStream_t stream) {
  const float* input = (const float*)d_in[0];
  const float* pk    = (const float*)d_in[1];
  const float* pv    = (const float*)d_in[2];
  const float* ln0_w = (const float*)d_in[3];
  const float* ln0_b = (const float*)d_in[4];
  const float* ln1_w = (const float*)d_in[5];
  const float* ln1_b = (const float*)d_in[6];
  const float* wqkv  = (const float*)d_in[7];
  const float* wo    = (const float*)d_in[8];
  const float* fc0_w = (const float*)d_in[9];
  const float* fc0_b = (const float*)d_in[10];
  const float* fc1_w = (const float*)d_in[11];
  const float* fc1_b = (const float*)d_in[12];
  float* out = (float*)d_out;

  float* ws   = (float*)d_ws;
  float* x_ln = ws;                    // 16*2048
  float* qkv  = x_ln + kB * kD;        // 16*6144
  float* attn = qkv + kB * 3 * kD;     // 16*2048
  float* hbuf = attn + kB * kD;        // 16*2048
  float* y_ln = hbuf + kB * kD;        // 16*2048
  float* act  = y_ln + kB * kD;        // 16*8192

  ln_rows_kernel<<<kB, 256, 0, stream>>>(input, ln0_w, ln0_b, x_ln);
  gemm16_wmma_kernel<<<(3 * kD) / 64, 128, 0, stream>>>(
      x_ln, wqkv, nullptr, nullptr, qkv, kD, 3 * kD, 0);
  attn_kernel<<<kB * kH, 256, 0, stream>>>(qkv, pk, pv, attn);
  gemm16_wmma_kernel<<<kD / 64, 128, 0, stream>>>(
      attn, wo, nullptr, input, hbuf, kD, kD, 1);
  ln_rows_kernel<<<kB, 256, 0, stream>>>(hbuf, ln1_w, ln1_b, y_ln);
  gemm16_wmma_kernel<<<kM / 64, 128, 0, stream>>>(
      y_ln, fc0_w, fc0_b, nullptr, act, kD, kM, 2);
  gemm16_wmma_kernel<<<kD / 64, 128, 0, stream>>>(
      act, fc1_w, fc1_b, hbuf, out, kM, kD, 3);
}